// GAT_80204219285522
// MI455X (gfx1250) — compile-verified
//
#include <hip/hip_runtime.h>

// ---------------------------------------------------------------------------
// GAT (3 layers) for gfx1250. GEMMs via V_WMMA_F32_16X16X4_F32 with an LDS-
// staged B panel (branch-free inner loop, 4 accumulators/wave). Edge softmax
// and aggregation via L2 atomics. Scratch in d_ws, attention staged in d_out.
// ---------------------------------------------------------------------------

#define GAT_N    20000
#define GAT_E    320000
#define GAT_IN   256
#define GAT_HID  64
#define GAT_NCLS 40

typedef __attribute__((ext_vector_type(2))) float v2f;
typedef __attribute__((ext_vector_type(8))) float v8f;

// ---------------- order-preserving float <-> u32 for atomicMax -------------
__device__ __forceinline__ unsigned f2ord(float f) {
  unsigned u = __float_as_uint(f);
  return (u & 0x80000000u) ? ~u : (u | 0x80000000u);
}
__device__ __forceinline__ float ord2f(unsigned o) {
  return __uint_as_float((o & 0x80000000u) ? (o & 0x7FFFFFFFu) : ~o);
}

// ---------------- WMMA f32 GEMM: Y[M,Nout] = X[M,K] @ W[K,Nout] ------------
// Block = 128 threads (4 waves) -> 64x64 C tile. Each wave: 16 rows x 64 cols
// via 4 accumulators. B panel (K x 64) staged in LDS, zero-padded past Nout,
// so the K loop is branch-free and EXEC is all-1s at every WMMA.
//
// A 16x4  : lanes 0-15 hold M=lane (K=+0,+1), lanes 16-31 hold M=lane-16 (K=+2,+3)
// B 4x16  : lanes 0-15 hold N=lane (K=+0,+1), lanes 16-31 hold N=lane-16 (K=+2,+3)
// C 16x16 : VGPR v -> row v (lanes 0-15) / row v+8 (lanes 16-31), col = lane&15
__global__ __launch_bounds__(128) void gat_gemm_wmma(
    const float* __restrict__ X, const float* __restrict__ W,
    float* __restrict__ Y, int M, int K, int Nout) {
  __shared__ float Bp[GAT_IN * 64];  // 256 x 64 floats = 64 KB

  const int tid  = threadIdx.x;
  const int lane = tid & 31;
  const int wave = tid >> 5;
  const int col0 = blockIdx.y * 64;
  const int row0 = blockIdx.x * 64 + wave * 16;

  // ---- cooperative stage of B panel into LDS (float4 chunks, zero-pad) ----
  for (int idx = tid; idx < GAT_IN * 16; idx += 128) {
    const int k  = idx >> 4;          // 0..K-1
    const int c4 = (idx & 15) << 2;   // 0..60, step 4
    const int gc = col0 + c4;
    float4 v = make_float4(0.f, 0.f, 0.f, 0.f);
    const float* Wr = W + (size_t)k * Nout;
    if (gc + 3 < Nout) {
      v = *reinterpret_cast<const float4*>(Wr + gc);
    } else {
      if (gc + 0 < Nout) v.x = Wr[gc + 0];
      if (gc + 1 < Nout) v.y = Wr[gc + 1];
      if (gc + 2 < Nout) v.z = Wr[gc + 2];
    }
    *reinterpret_cast<float4*>(&Bp[k * 64 + c4]) = v;
  }
  __syncthreads();

  const int half = lane >> 4;   // 0: lanes 0-15, 1: lanes 16-31
  const int l16  = lane & 15;
  int arow = row0 + l16;
  arow = (arow < M) ? arow : (M - 1);            // clamp: keeps loads in-bounds,
  const float* __restrict__ Xr = X + (size_t)arow * K;  // EXEC stays full

  v8f acc0 = {}, acc1 = {}, acc2 = {}, acc3 = {};
  for (int kk = 0; kk < K; kk += 4) {
    const int ka = kk + 2 * half;
    v2f a;
    a.x = Xr[ka];
    a.y = Xr[ka + 1];
    const float* b0 = &Bp[(ka    ) * 64 + l16];
    const float* b1 = &Bp[(ka + 1) * 64 + l16];
    v2f b;
    b.x = b0[0];  b.y = b1[0];
    acc0 = __builtin_amdgcn_wmma_f32_16x16x4_f32(false, a, false, b, (short)0, acc0, false, false);
    b.x = b0[16]; b.y = b1[16];
    acc1 = __builtin_amdgcn_wmma_f32_16x16x4_f32(false, a, false, b, (short)0, acc1, false, false);
    b.x = b0[32]; b.y = b1[32];
    acc2 = __builtin_amdgcn_wmma_f32_16x16x4_f32(false, a, false, b, (short)0, acc2, false, false);
    b.x = b0[48]; b.y = b1[48];
    acc3 = __builtin_amdgcn_wmma_f32_16x16x4_f32(false, a, false, b, (short)0, acc3, false, false);
  }

  // ---- store 4 tiles, guarded ----
  const int crow0 = row0 + half * 8;
  v8f accs[4] = {acc0, acc1, acc2, acc3};
#pragma unroll
  for (int t = 0; t < 4; ++t) {
    const int ccol = col0 + t * 16 + l16;
    if (ccol < Nout) {
#pragma unroll
      for (int v = 0; v < 8; ++v) {
        const int r = crow0 + v;
        if (r < M) Y[(size_t)r * Nout + ccol] = accs[t][v];
      }
    }
  }
}

// ---------------- per-node attention logits: el/er [N,H] -------------------
__global__ void gat_node_lr(const float* __restrict__ feat,
                            const float* __restrict__ al,
                            const float* __restrict__ ar,
                            float* __restrict__ el, float* __restrict__ er,
                            int N, int H, int D) {
  int n = blockIdx.x * blockDim.x + threadIdx.x;
  if (n >= N) return;
  for (int h = 0; h < H; ++h) {
    const float* f  = feat + (size_t)n * H * D + (size_t)h * D;
    const float* pl = al + (size_t)h * D;
    const float* pr = ar + (size_t)h * D;
    float sl = 0.0f, sr = 0.0f;
    for (int d = 0; d < D; ++d) {
      float v = f[d];
      sl += v * pl[d];
      sr += v * pr[d];
    }
    el[n * H + h] = sl;
    er[n * H + h] = sr;
  }
}

// ---------------- edge score + LeakyReLU + segment-max ---------------------
__global__ void gat_edge_score(const float* __restrict__ el,
                               const float* __restrict__ er,
                               const int* __restrict__ src,
                               const int* __restrict__ dst,
                               float* __restrict__ ebuf,
                               unsigned* __restrict__ mord, int E, int H) {
  int t = blockIdx.x * blockDim.x + threadIdx.x;
  if (t >= E * H) return;
  int e = t / H, h = t - e * H;
  int s = src[e], d = dst[e];
  float v = el[s * H + h] + er[d * H + h];
  v = (v > 0.0f) ? v : 0.2f * v;  // LeakyReLU(0.2)
  ebuf[t] = v;
  atomicMax(&mord[d * H + h], f2ord(v));
}

// ---------------- exp(e - max) + segment-sum -------------------------------
__global__ void gat_edge_exp(float* __restrict__ ebuf,
                             const unsigned* __restrict__ mord,
                             const int* __restrict__ dst,
                             float* __restrict__ denom, int E, int H) {
  int t = blockIdx.x * blockDim.x + threadIdx.x;
  if (t >= E * H) return;
  int e = t / H, h = t - e * H;
  int d = dst[e];
  float m  = ord2f(mord[d * H + h]);
  float ex = __expf(ebuf[t] - m);
  ebuf[t] = ex;
  atomicAdd(&denom[d * H + h], ex);
}

// ---------------- normalize: a = ex / denom[dst] (final attention) ---------
__global__ void gat_edge_norm(float* __restrict__ ebuf,
                              const float* __restrict__ denom,
                              const int* __restrict__ dst, int E, int H) {
  int t = blockIdx.x * blockDim.x + threadIdx.x;
  if (t >= E * H) return;
  int e = t / H, h = t - e * H;
  ebuf[t] = ebuf[t] / denom[dst[e] * H + h];
}

// ---------------- message aggregation: out[dst] += feat[src] * a -----------
__global__ void gat_edge_msg(const float* __restrict__ feat,
                             const float* __restrict__ attn,
                             const int* __restrict__ src,
                             const int* __restrict__ dst,
                             float* __restrict__ accum,
                             long long total, int H, int D) {
  long long t = (long long)blockIdx.x * blockDim.x + threadIdx.x;
  if (t >= total) return;
  const int HD = H * D;
  int e = (int)(t / HD);
  int r = (int)(t - (long long)e * HD);
  float a = attn[(size_t)e * H + r / D];
  atomicAdd(&accum[(size_t)dst[e] * HD + r],
            feat[(size_t)src[e] * HD + r] * a);
}

// ---------------- bias add (optionally out-of-place) -----------------------
__global__ void gat_bias(float* __restrict__ out, const float* __restrict__ in,
                         const float* __restrict__ b, long long total, int F) {
  long long t = (long long)blockIdx.x * blockDim.x + threadIdx.x;
  if (t >= total) return;
  out[t] = in[t] + b[(int)(t % F)];
}

// ---------------------------------------------------------------------------
static void run_gat_layer(const float* X, const float* W, const float* al,
                          const float* ar, const float* b, const int* src,
                          const int* dst, float* feat, float* accum,
                          float* xout, float* attn, float* el, float* er,
                          unsigned* mo, float* denom, int H, int D,
                          hipStream_t stream) {
  const int N = GAT_N, E = GAT_E, K = GAT_IN;
  const int Fout = H * D;

  // zero accumulators / softmax state (graph-capture-safe async memsets)
  hipMemsetAsync(accum, 0, (size_t)N * Fout * sizeof(float), stream);
  hipMemsetAsync(mo,    0, (size_t)N * H * sizeof(unsigned), stream);  // 0 < any f2ord
  hipMemsetAsync(denom, 0, (size_t)N * H * sizeof(float), stream);

  // feat = X @ W   (WMMA, 64x64 tile per block)
  dim3 ggrid((N + 63) / 64, (Fout + 63) / 64);
  gat_gemm_wmma<<<ggrid, 128, 0, stream>>>(X, W, feat, N, K, Fout);

  // el / er
  gat_node_lr<<<(N + 255) / 256, 256, 0, stream>>>(feat, al, ar, el, er, N, H, D);

  // edge softmax (e -> exp -> a), staged in-place in the attn output buffer
  int eh = E * H;
  gat_edge_score<<<(eh + 255) / 256, 256, 0, stream>>>(el, er, src, dst, attn, mo, E, H);
  gat_edge_exp  <<<(eh + 255) / 256, 256, 0, stream>>>(attn, mo, dst, denom, E, H);
  gat_edge_norm <<<(eh + 255) / 256, 256, 0, stream>>>(attn, denom, dst, E, H);

  // aggregation
  long long tot = (long long)E * Fout;
  gat_edge_msg<<<(unsigned)((tot + 255) / 256), 256, 0, stream>>>(
      feat, attn, src, dst, accum, tot, H, D);

  // bias (xout may alias accum)
  long long nb = (long long)N * Fout;
  gat_bias<<<(unsigned)((nb + 255) / 256), 256, 0, stream>>>(xout, accum, b, nb, Fout);
}

extern "C" void kernel_launch(void* const* d_in, const int* in_sizes, int n_in,
                              void* d_out, int out_size, void* d_ws, size_t ws_size,
                              hipStream_t stream) {
  (void)in_sizes; (void)n_in; (void)out_size; (void)ws_size;

  const float* h   = (const float*)d_in[0];
  const int*   src = (const int*)  d_in[1];
  const int*   dst = (const int*)  d_in[2];
  const float* W1  = (const float*)d_in[3];
  const float* al1 = (const float*)d_in[4];
  const float* ar1 = (const float*)d_in[5];
  const float* b1  = (const float*)d_in[6];
  const float* W2  = (const float*)d_in[7];
  const float* al2 = (const float*)d_in[8];
  const float* ar2 = (const float*)d_in[9];
  const float* b2  = (const float*)d_in[10];
  const float* W3  = (const float*)d_in[11];
  const float* al3 = (const float*)d_in[12];
  const float* ar3 = (const float*)d_in[13];
  const float* b3  = (const float*)d_in[14];

  const size_t N = GAT_N, E = GAT_E;

  // workspace layout
  float*    ws    = (float*)d_ws;
  float*    bufA  = ws;                    // feat1, later feat3
  float*    bufB  = bufA + N * 256;        // accum1 -> x1
  float*    bufC  = bufB + N * 256;        // feat2, later accum3
  float*    bufD  = bufC + N * 256;        // accum2 -> x2
  float*    el    = bufD + N * 256;
  float*    er    = el + N * 4;
  unsigned* mo    = (unsigned*)(er + N * 4);
  float*    denom = (float*)(mo + N * 4);

  // output layout: logits [N,40] | attn1 [E,4] | attn2 [E,4] | attn3 [E,1]
  float* logits = (float*)d_out;
  float* attn1  = logits + N * GAT_NCLS;
  float* attn2  = attn1 + E * 4;
  float* attn3  = attn2 + E * 4;

  // Layer 1: X=h [N,256] -> feat bufA, accum/x1 bufB   (H=4, D=64)
  run_gat_layer(h, W1, al1, ar1, b1, src, dst,
                bufA, bufB, bufB, attn1, el, er, mo, denom, 4, 64, stream);

  // Layer 2: X=bufB -> feat bufC, accum/x2 bufD        (H=4, D=64)
  run_gat_layer(bufB, W2, al2, ar2, b2, src, dst,
                bufC, bufD, bufD, attn2, el, er, mo, denom, 4, 64, stream);

  // Layer 3: X=bufD -> feat bufA, accum bufC, logits   (H=1, D=40)
  run_gat_layer(bufD, W3, al3, ar3, b3, src, dst,
                bufA, bufC, logits, attn3, el, er, mo, denom, 1, GAT_NCLS, stream);
}